// SoftMoE_46789373722624
// MI455X (gfx1250) — compile-verified
//
#include <hip/hip_runtime.h>
#include <hip/hip_bf16.h>

// ---------------- problem constants ----------------
constexpr int B   = 8;
constexpr int S   = 4096;
constexpr int D   = 768;
constexpr int M   = 64;
constexpr int FFN = 3072;
constexpr int HD  = 64;
constexpr int H   = D / HD;   // 12
constexpr float EPS = 1e-5f;

static_assert(D == 3 * 256, "prenorm assumes 3 elems/thread at 256 threads");

typedef __attribute__((ext_vector_type(2))) float v2f;
typedef __attribute__((ext_vector_type(8))) float v8f;
typedef __attribute__((ext_vector_type(4))) unsigned int v4u;
typedef __attribute__((ext_vector_type(8))) int v8i;
typedef __attribute__((ext_vector_type(4))) int v4i;

// D = A(16x4) * B(4x16) + C  -- fp32 WMMA, wave32
__device__ inline v8f wmma4(v2f a, v2f b, v8f c) {
    return __builtin_amdgcn_wmma_f32_16x16x4_f32(
        /*neg_a=*/false, a, /*neg_b=*/false, b,
        /*c_mod=*/(short)0, c, /*reuse_a=*/false, /*reuse_b=*/false);
}

// Fragment loads from LDS per ISA §7.12.2 layouts.
// Matrix stored [row][k] (A matrix, or B stored transposed as [n][k]):
//   element(row0+lane%16, k0 + v + 2*(lane/16))
__device__ inline v2f frag_rk(const float* base, int stride, int row0, int k0, int lane) {
    int lm = lane & 15, lh = lane >> 4;
    const float* p = base + (size_t)(row0 + lm) * stride + k0 + 2 * lh;
    v2f r; r.x = p[0]; r.y = p[1];
    return r;
}
// Matrix stored [k][n] (natural B layout): element(k0 + v + 2*(lane/16), n0+lane%16)
__device__ inline v2f frag_kn(const float* base, int stride, int k0, int n0, int lane) {
    int lm = lane & 15, lh = lane >> 4;
    const float* p = base + (size_t)(k0 + 2 * lh) * stride + n0 + lm;
    v2f r; r.x = p[0]; r.y = p[stride];
    return r;
}

__device__ inline float block_sum256(float v, float* red) {
    #pragma unroll
    for (int m = 16; m; m >>= 1) v += __shfl_xor(v, m, 32);
    __syncthreads();
    if ((threadIdx.x & 31) == 0) red[threadIdx.x >> 5] = v;
    __syncthreads();
    float s = 0.f;
    #pragma unroll
    for (int i = 0; i < 8; ++i) s += red[i];
    return s;
}

// ---------------- TDM: async 2-D f32 tile load (global -> LDS) ----------------
// Builds a D# per CDNA5 ISA §8.3/8.4 and issues TENSOR_LOAD_TO_LDS.
//   tile_x x tile_y f32 elements; row pitch = stride_x elems; LDS gets
//   pad_amount(4 DW) inserted every pad_interval(128 DW) -> LDS row stride 132.
__device__ inline unsigned lds_off_u32(const void* p) {
    // low 32 bits of a generic LDS pointer are the LDS byte offset
    return (unsigned)(size_t)p;
}

__device__ inline void tdm_load_tile_f32(const float* gptr, unsigned lds_addr,
                                         unsigned tile_x, unsigned tile_y,
                                         unsigned tensor_x, unsigned tensor_y,
                                         unsigned long long stride_x,
                                         unsigned pad_int_code, unsigned pad_amt_code) {
    unsigned long long ga = (unsigned long long)(size_t)gptr;
    v4u g0;
    g0.x = 1u;                                             // count=1 (valid), no gather
    g0.y = lds_addr;                                       // lds_addr [63:32]
    g0.z = (unsigned)(ga & 0xffffffffull);                 // global_addr lo
    g0.w = (unsigned)((ga >> 32) & 0x01ffffffull)          // global_addr [56:32]
         | (2u << 30);                                     // type = 2 ("image")
    v8i g1;
    g1[0] = (int)((2u << 16)                               // data_size = 4B
                | (1u << 20)                               // pad_enable
                | (pad_int_code << 22)                     // pad_interval
                | (pad_amt_code << 25));                   // pad_amount
    g1[1] = (int)((tensor_x & 0xffffu) << 16);             // tensor_dim0 lo16 @ [63:48]
    g1[2] = (int)((tensor_x >> 16) | ((tensor_y & 0xffffu) << 16));
    g1[3] = (int)((tensor_y >> 16) | (tile_x << 16));      // tile_dim0 @ [127:112]
    g1[4] = (int)tile_y;                                   // tile_dim1 (tile_dim2 = 0)
    g1[5] = (int)(stride_x & 0xffffffffull);               // tensor_dim0_stride lo32
    g1[6] = (int)((stride_x >> 32) & 0xffffull);           // stride hi16; dim1_stride = 0
    g1[7] = 0;
    v4i z4 = {0, 0, 0, 0};                                 // 2-D tensor: groups 2/3 unused
    v8i z8 = {0, 0, 0, 0, 0, 0, 0, 0};
    __builtin_amdgcn_tensor_load_to_lds(g0, g1, z4, z4, z8, 0);
}

// ---------------- 1) fused RMSNorm + L2 norm of tokens ----------------
__global__ __launch_bounds__(256) void k_prenorm(const float* __restrict__ x,
                                                 const float* __restrict__ nw,
                                                 float* __restrict__ xn,
                                                 float* __restrict__ nt) {
    __shared__ float red[8];
    const int tok = blockIdx.x;           // b*S + s
    const int t = threadIdx.x;
    const float* xi = x + (size_t)tok * D;
    float v[3], ss = 0.f;
    #pragma unroll
    for (int i = 0; i < 3; ++i) { v[i] = xi[t + 256 * i]; ss += v[i] * v[i]; }
    ss = block_sum256(ss, red);
    const float r = rsqrtf(ss * (1.f / D) + EPS);
    float yv[3], ss2 = 0.f;
    #pragma unroll
    for (int i = 0; i < 3; ++i) {
        yv[i] = v[i] * r * nw[t + 256 * i];
        ss2 += yv[i] * yv[i];
    }
    float* xo = xn + (size_t)tok * D;
    #pragma unroll
    for (int i = 0; i < 3; ++i) xo[t + 256 * i] = yv[i];
    ss2 = block_sum256(ss2, red);
    const float inv = 1.f / fmaxf(sqrtf(ss2), EPS);
    float* no = nt + (size_t)tok * D;
    #pragma unroll
    for (int i = 0; i < 3; ++i) no[t + 256 * i] = yv[i] * inv;
}

// ---------------- 2) slots: l2norm * scale ----------------
__global__ __launch_bounds__(256) void k_slotnorm(const float* __restrict__ slots,
                                                  const float* __restrict__ scale,
                                                  float* __restrict__ sl) {
    __shared__ float red[8];
    const int m = blockIdx.x, t = threadIdx.x;
    const float* si = slots + (size_t)m * D;
    float v[3], ss = 0.f;
    #pragma unroll
    for (int i = 0; i < 3; ++i) { v[i] = si[t + 256 * i]; ss += v[i] * v[i]; }
    ss = block_sum256(ss, red);
    const float inv = scale[m] / fmaxf(sqrtf(ss), EPS);
    float* so = sl + (size_t)m * D;
    #pragma unroll
    for (int i = 0; i < 3; ++i) so[t + 256 * i] = v[i] * inv;
}

// ---------------- 3) logits[bh,m,s] = slh . nth  (K = HD = 64) ----------------
__global__ __launch_bounds__(256) void k_logits(const float* __restrict__ sl,
                                                const float* __restrict__ nt,
                                                float* __restrict__ logits) {
    constexpr int AST = 68, BST = 68;
    __shared__ float As[64 * AST];    // slh: [m][hd]
    __shared__ float Bs[128 * BST];   // nth tile, stored [s][hd]  (== B^T layout [n][k])
    const int s0 = blockIdx.x * 128;
    const int bh = blockIdx.y;
    const int b = bh / H, h = bh % H;
    const int t = threadIdx.x;
    for (int i = t; i < 64 * 64; i += 256) {
        int m = i >> 6, hd = i & 63;
        As[m * AST + hd] = sl[(size_t)m * D + h * HD + hd];
    }
    for (int i = t; i < 128 * 64; i += 256) {
        int sr = i >> 6, hd = i & 63;
        Bs[sr * BST + hd] = nt[((size_t)b * S + s0 + sr) * D + h * HD + hd];
    }
    __syncthreads();
    const int wave = t >> 5, lane = t & 31;
    const int mt = wave >> 1;              // 0..3  (m tile)
    const int stbase = (wave & 1) * 4;     // s tiles: 4 per wave
    float* lout = logits + (size_t)bh * M * S;
    const int lm = lane & 15, lh = lane >> 4;
    for (int j = 0; j < 4; ++j) {
        const int st = stbase + j;
        v8f acc = {};
        #pragma unroll
        for (int k0 = 0; k0 < 64; k0 += 4) {
            v2f a  = frag_rk(As, AST, mt * 16, k0, lane);
            v2f bf = frag_rk(Bs, BST, st * 16, k0, lane);   // [n][k] layout
            acc = wmma4(a, bf, acc);
        }
        const int scol = s0 + st * 16 + lm;
        #pragma unroll
        for (int v = 0; v < 8; ++v) {
            const int mm = mt * 16 + v + 8 * lh;
            lout[(size_t)mm * S + scol] = acc[v];
        }
    }
}

// ---------------- 4) dispatch: y[b,m,:] = softmax_s(logits) @ xh  (K = S) -------
__global__ __launch_bounds__(256) void k_dispatch(const float* __restrict__ logits,
                                                  const float* __restrict__ xn,
                                                  float* __restrict__ y) {
    constexpr int PST = 68, XST = 68;
    __shared__ float Ps[64 * PST];     // probs chunk [m][s_local]
    __shared__ float Xs[64 * XST];     // xh chunk [s_local][hd]  ([k][n] layout)
    __shared__ float rmax[64], rsinv[64];
    const int bh = blockIdx.x;
    const int b = bh / H, h = bh % H;
    const int t = threadIdx.x, wave = t >> 5, lane = t & 31;
    const float* lg = logits + (size_t)bh * M * S;
    // pass 1: per-row max and sum of exp over S (rows are L2-hot, 1MB slab)
    for (int j = 0; j < 8; ++j) {
        const int r = wave * 8 + j;
        const float* row = lg + (size_t)r * S;
        float mx = -3.4e38f;
        for (int s = lane; s < S; s += 32) mx = fmaxf(mx, row[s]);
        #pragma unroll
        for (int msk = 16; msk; msk >>= 1) mx = fmaxf(mx, __shfl_xor(mx, msk, 32));
        float sm = 0.f;
        for (int s = lane; s < S; s += 32) sm += expf(row[s] - mx);
        #pragma unroll
        for (int msk = 16; msk; msk >>= 1) sm += __shfl_xor(sm, msk, 32);
        if (lane == 0) { rmax[r] = mx; rsinv[r] = 1.f / sm; }
    }
    const int mt = wave >> 1;               // m tile 0..3
    const int htbase = (wave & 1) * 2;      // 2 hd tiles per wave
    v8f acc0 = {}, acc1 = {};
    for (int sc = 0; sc < S; sc += 64) {
        __syncthreads();
        for (int i = t; i < 64 * 64; i += 256) {
            int m = i >> 6, sl_ = i & 63;
            Ps[m * PST + sl_] = expf(lg[(size_t)m * S + sc + sl_] - rmax[m]) * rsinv[m];
        }
        for (int i = t; i < 64 * 64; i += 256) {
            int sr = i >> 6, hd = i & 63;
            Xs[sr * XST + hd] = xn[((size_t)b * S + sc + sr) * D + h * HD + hd];
        }
        __syncthreads();
        #pragma unroll
        for (int k0 = 0; k0 < 64; k0 += 4) {
            v2f a  = frag_rk(Ps, PST, mt * 16, k0, lane);
            v2f b0 = frag_kn(Xs, XST, k0, (htbase + 0) * 16, lane);
            v2f b1 = frag_kn(Xs, XST, k0, (htbase + 1) * 16, lane);
            acc0 = wmma4(a, b0, acc0);
            acc1 = wmma4(a, b1, acc1);
        }
    }
    const int lm = lane & 15, lh = lane >> 4;
    #pragma unroll
    for (int v = 0; v < 8; ++v) {
        const int mm = mt * 16 + v + 8 * lh;
        const size_t base = ((size_t)b * M + mm) * D + h * HD;
        y[base + (htbase + 0) * 16 + lm] = acc0[v];
        y[base + (htbase + 1) * 16 + lm] = acc1[v];
    }
}

// ---------------- 5) expert fc1 + exact GELU  (per expert: [8 x FFN], K = D) ----
// Weight tiles stream global->LDS via the Tensor Data Mover, double buffered;
// waves do only ds_load + v_wmma.  LDS row stride 132 comes from TDM pad
// (4 DW pad every 128 DW).
__global__ __launch_bounds__(256) void k_fc1(const float* __restrict__ y,
                                             const float* __restrict__ w,
                                             const float* __restrict__ bias,
                                             float* __restrict__ hid) {
    constexpr int AST = 36, BST = 132;
    __shared__ float As[16 * AST];      // [16 rows (8 valid)][Kc=32]
    __shared__ float Bs[2][32 * BST];   // double-buffered [Kc=32][128] (+pad)
    const int m  = blockIdx.y;
    const int f0 = blockIdx.x * 128;
    const int t = threadIdx.x, wave = t >> 5, lane = t & 31;
    const int n0 = wave * 16;
    const float* wbase = w + (size_t)m * D * FFN + f0;
    constexpr int NCH = D / 32;         // 24 K-chunks
    if (wave == 0)
        tdm_load_tile_f32(wbase, lds_off_u32(&Bs[0][0]), 128, 32, FFN, D,
                          (unsigned long long)FFN, 6u, 3u);
    v8f acc = {};
    for (int c = 0; c < NCH; ++c) {
        const int d0 = c * 32;
        for (int i = t; i < 16 * 32; i += 256) {
            int r = i >> 5, k = i & 31;
            As[r * AST + k] = (r < B) ? y[((size_t)r * M + m) * D + d0 + k] : 0.f;
        }
        if (wave == 0) {
            if (c + 1 < NCH) {
                tdm_load_tile_f32(wbase + (size_t)(d0 + 32) * FFN,
                                  lds_off_u32(&Bs[(c + 1) & 1][0]), 128, 32, FFN, D,
                                  (unsigned long long)FFN, 6u, 3u);
                __builtin_amdgcn_s_wait_tensorcnt(1);
            } else {
                __builtin_amdgcn_s_wait_tensorcnt(0);
            }
        }
        __syncthreads();
        const float* Bc = &Bs[c & 1][0];
        #pragma unroll
        for (int k0 = 0; k0 < 32; k0 += 4) {
            v2f a  = frag_rk(As, AST, 0, k0, lane);
            v2f bf = frag_kn(Bc, BST, k0, n0, lane);
            acc = wmma4(a, bf, acc);
        }
        __syncthreads();
    }
    const int lm = lane & 15, lh = lane >> 4;
    if (lh == 0) {                       // rows 0..7 are the valid batch rows
        const int f = f0 + n0 + lm;
        const float bv = bias[(size_t)m * FFN + f];
        #pragma unroll
        for (int v = 0; v < 8; ++v) {    // v = batch index
            float val = acc[v] + bv;
            val = 0.5f * val * (1.f + erff(val * 0.70710678118654752f));
            hid[((size_t)v * M + m) * FFN + f] = val;
        }
    }
}

// ---------------- 6) expert fc2  (per expert: [8 x D], K = FFN) ----------------
__global__ __launch_bounds__(256) void k_fc2(const float* __restrict__ hid,
                                             const float* __restrict__ w,
                                             const float* __restrict__ bias,
                                             float* __restrict__ y2) {
    constexpr int AST = 36, BST = 132;
    __shared__ float As[16 * AST];
    __shared__ float Bs[2][32 * BST];
    const int m  = blockIdx.y;
    const int d0 = blockIdx.x * 128;
    const int t = threadIdx.x, wave = t >> 5, lane = t & 31;
    const int n0 = wave * 16;
    const float* wbase = w + (size_t)m * FFN * D + d0;
    constexpr int NCH = FFN / 32;       // 96 K-chunks
    if (wave == 0)
        tdm_load_tile_f32(wbase, lds_off_u32(&Bs[0][0]), 128, 32, D, FFN,
                          (unsigned long long)D, 6u, 3u);
    v8f acc = {};
    for (int c = 0; c < NCH; ++c) {
        const int f0 = c * 32;
        for (int i = t; i < 16 * 32; i += 256) {
            int r = i >> 5, k = i & 31;
            As[r * AST + k] = (r < B) ? hid[((size_t)r * M + m) * FFN + f0 + k] : 0.f;
        }
        if (wave == 0) {
            if (c + 1 < NCH) {
                tdm_load_tile_f32(wbase + (size_t)(f0 + 32) * D,
                                  lds_off_u32(&Bs[(c + 1) & 1][0]), 128, 32, D, FFN,
                                  (unsigned long long)D, 6u, 3u);
                __builtin_amdgcn_s_wait_tensorcnt(1);
            } else {
                __builtin_amdgcn_s_wait_tensorcnt(0);
            }
        }
        __syncthreads();
        const float* Bc = &Bs[c & 1][0];
        #pragma unroll
        for (int k0 = 0; k0 < 32; k0 += 4) {
            v2f a  = frag_rk(As, AST, 0, k0, lane);
            v2f bf = frag_kn(Bc, BST, k0, n0, lane);
            acc = wmma4(a, bf, acc);
        }
        __syncthreads();
    }
    const int lm = lane & 15, lh = lane >> 4;
    if (lh == 0) {
        const int d = d0 + n0 + lm;
        const float bv = bias[(size_t)m * D + d];
        #pragma unroll
        for (int v = 0; v < 8; ++v)
            y2[((size_t)v * M + m) * D + d] = acc[v] + bv;
    }
}

// ---------------- 7) combine: out = softmax_m(logits^T) @ y2h  (K = M = 64) -----
__global__ __launch_bounds__(128) void k_combine(const float* __restrict__ logits,
                                                 const float* __restrict__ y2,
                                                 float* __restrict__ out) {
    constexpr int PST = 65, YST = 68;
    __shared__ float Ps[128 * PST];   // P2 [s_local][m]   (A: [row=s][k=m])
    __shared__ float Ys[64 * YST];    // y2h [m][hd]       (B: [k][n])
    const int s0 = blockIdx.x * 128;
    const int bh = blockIdx.y;
    const int b = bh / H, h = bh % H;
    const int t = threadIdx.x, wave = t >> 5, lane = t & 31;
    for (int i = t; i < 64 * 64; i += 128) {
        int mr = i >> 6, hd = i & 63;
        Ys[mr * YST + hd] = y2[((size_t)b * M + mr) * D + h * HD + hd];
    }
    // softmax over m for this thread's token s = s0 + t (coalesced across threads)
    const float* lg = logits + (size_t)bh * M * S;
    {
        float mx = -3.4e38f;
        for (int mr = 0; mr < M; ++mr) {
            float v = lg[(size_t)mr * S + s0 + t];
            Ps[t * PST + mr] = v;
            mx = fmaxf(mx, v);
        }
        float sm = 0.f;
        for (int mr = 0; mr < M; ++mr) {
            float e = expf(Ps[t * PST + mr] - mx);
            Ps[t * PST + mr] = e;
            sm += e;
        }
        const float inv = 1.f / sm;
        for (int mr = 0; mr < M; ++mr) Ps[t * PST + mr] *= inv;
    }
    __syncthreads();
    const int lm = lane & 15, lh = lane >> 4;
    for (int j = 0; j < 2; ++j) {
        const int st = wave * 2 + j;            // 8 s-tiles over 4 waves
        for (int ht = 0; ht < 4; ++ht) {
            v8f acc = {};
            #pragma unroll
            for (int k0 = 0; k0 < 64; k0 += 4) {
                v2f a  = frag_rk(Ps, PST, st * 16, k0, lane);
                v2f bf = frag_kn(Ys, YST, k0, ht * 16, lane);
                acc = wmma4(a, bf, acc);
            }
            const int dcol = h * HD + ht * 16 + lm;
            #pragma unroll
            for (int v = 0; v < 8; ++v) {
                const int ss_ = s0 + st * 16 + v + 8 * lh;
                out[((size_t)b * S + ss_) * D + dcol] = acc[v];
            }
        }
    }
}

// ---------------- host launcher ----------------
extern "C" void kernel_launch(void* const* d_in, const int* in_sizes, int n_in,
                              void* d_out, int out_size, void* d_ws, size_t ws_size,
                              hipStream_t stream) {
    (void)in_sizes; (void)n_in; (void)out_size; (void)ws_size;
    const float* x      = (const float*)d_in[0];
    const float* slots  = (const float*)d_in[1];
    const float* scale  = (const float*)d_in[2];
    const float* fc1_w  = (const float*)d_in[3];
    const float* fc1_b  = (const float*)d_in[4];
    const float* fc2_w  = (const float*)d_in[5];
    const float* fc2_b  = (const float*)d_in[6];
    const float* norm_w = (const float*)d_in[7];
    float* out = (float*)d_out;

    float* ws     = (float*)d_ws;
    float* xn     = ws;                                    // B*S*D
    float* nt     = xn     + (size_t)B * S * D;            // B*S*D
    float* sl     = nt     + (size_t)B * S * D;            // M*D
    float* logits = sl     + (size_t)M * D;                // B*H*M*S
    float* y      = logits + (size_t)B * H * M * S;        // B*M*D
    float* hid    = y      + (size_t)B * M * D;            // B*M*FFN
    float* y2     = hid    + (size_t)B * M * FFN;          // B*M*D

    k_prenorm <<<B * S, 256, 0, stream>>>(x, norm_w, xn, nt);
    k_slotnorm<<<M, 256, 0, stream>>>(slots, scale, sl);
    k_logits  <<<dim3(S / 128, B * H), 256, 0, stream>>>(sl, nt, logits);
    k_dispatch<<<B * H, 256, 0, stream>>>(logits, xn, y);
    k_fc1     <<<dim3(FFN / 128, M), 256, 0, stream>>>(y, fc1_w, fc1_b, hid);
    k_fc2     <<<dim3(D / 128, M), 256, 0, stream>>>(hid, fc2_w, fc2_b, y2);
    k_combine <<<dim3(S / 128, B * H), 128, 0, stream>>>(logits, y2, out);
}